// HouseholdAssignmentGNN_43310450213611
// MI455X (gfx1250) — compile-verified
//
#include <hip/hip_runtime.h>

typedef __attribute__((ext_vector_type(2))) float v2f;
typedef __attribute__((ext_vector_type(8))) float v8f;

// ---------------------------------------------------------------------------
// Zero scratch accumulators (agg1 | agg2 | cnt laid out contiguously).
// ---------------------------------------------------------------------------
__global__ void zero_f32(float* __restrict__ p, long long n) {
  long long i = (long long)blockIdx.x * blockDim.x + threadIdx.x;
  long long stride = (long long)gridDim.x * blockDim.x;
  for (; i < n; i += stride) p[i] = 0.0f;
}

// ---------------------------------------------------------------------------
// Edge scatter, layer 1: per edge add x[src][0..15] into agg1[dst], count++.
// fp32 global atomics; destinations (<13MB) are L2-resident on MI455X.
// ---------------------------------------------------------------------------
__global__ void scatter_deg_feat16(const float* __restrict__ x,
                                   const long long* __restrict__ src,
                                   const long long* __restrict__ dst,
                                   float* __restrict__ agg,
                                   float* __restrict__ cnt,
                                   long long n_edges) {
  long long e = (long long)blockIdx.x * blockDim.x + threadIdx.x;
  if (e >= n_edges) return;
  const int s = (int)src[e];
  const int d = (int)dst[e];
  atomicAdd(&cnt[d], 1.0f);
  const float4* xs = (const float4*)(x + (long long)s * 16);
  float* ad = agg + (long long)d * 16;
#pragma unroll
  for (int v = 0; v < 4; ++v) {
    float4 f = xs[v];
    atomicAdd(&ad[v * 4 + 0], f.x);
    atomicAdd(&ad[v * 4 + 1], f.y);
    atomicAdd(&ad[v * 4 + 2], f.z);
    atomicAdd(&ad[v * 4 + 3], f.w);
  }
}

// ---------------------------------------------------------------------------
// Edge scatter, layer 2: per edge add h1[src][0..31] into agg2[dst].
// ---------------------------------------------------------------------------
__global__ void scatter_feat32(const float* __restrict__ h,
                               const long long* __restrict__ src,
                               const long long* __restrict__ dst,
                               float* __restrict__ agg,
                               long long n_edges) {
  long long e = (long long)blockIdx.x * blockDim.x + threadIdx.x;
  if (e >= n_edges) return;
  const int s = (int)src[e];
  const int d = (int)dst[e];
  const float4* hs = (const float4*)(h + (long long)s * 32);
  float* ad = agg + (long long)d * 32;
#pragma unroll
  for (int v = 0; v < 8; ++v) {
    float4 f = hs[v];
    atomicAdd(&ad[v * 4 + 0], f.x);
    atomicAdd(&ad[v * 4 + 1], f.y);
    atomicAdd(&ad[v * 4 + 2], f.z);
    atomicAdd(&ad[v * 4 + 3], f.w);
  }
}

__global__ void inv_count(const float* __restrict__ cnt,
                          float* __restrict__ invc, int n) {
  int i = blockIdx.x * blockDim.x + threadIdx.x;
  if (i < n) invc[i] = 1.0f / fmaxf(cnt[i], 1.0f);
}

// ---------------------------------------------------------------------------
// SAGE layer: out = relu( (agg*invc) @ wl + hself @ wr + bias ), fp32 WMMA.
// One wave owns one 16-node row block; both A tiles (mean & self) live in
// VGPRs and are reused across the OUTC/16 N-tiles. The uniform partial-row
// check is hoisted OUTSIDE the tile loop so the hot path stores are
// unconditional (no EXEC save/restore).
// A (16x4 f32): m = lane&15, k pair = (lane>>4)*2.
// B (4x16 f32): n = lane&15, k pair = (lane>>4)*2.
// C/D (16x16 f32): VGPR r -> M = r + (lane>>4)*8, N = lane&15.
// ---------------------------------------------------------------------------
template <int K, int OUTC>
__global__ __launch_bounds__(256) void sage_layer_wmma(
    const float* __restrict__ agg, const float* __restrict__ invc,
    const float* __restrict__ hself, const float* __restrict__ wl,
    const float* __restrict__ wr, const float* __restrict__ bias,
    float* __restrict__ out, int n_nodes) {
  const int lane = threadIdx.x & 31;
  const int wave = (blockIdx.x * blockDim.x + threadIdx.x) >> 5;
  const int mtiles = (n_nodes + 15) >> 4;
  if (wave >= mtiles) return;  // uniform per wave: EXEC all-1s for WMMA

  const int m = (wave << 4) + (lane & 15);
  const int mc = (m < n_nodes) ? m : (n_nodes - 1);  // clamped (safe) load idx
  const int kb = (lane >> 4) << 1;                   // 0 or 2
  constexpr int KS = K / 4;

  const float ic = invc[mc];
  const v2f* aggp = (const v2f*)(agg + (long long)mc * K);
  const v2f* selp = (const v2f*)(hself + (long long)mc * K);
  v2f aMean[KS], aSelf[KS];
#pragma unroll
  for (int s = 0; s < KS; ++s) {
    const int kp = s * 2 + (kb >> 1);
    v2f am = aggp[kp];
    aMean[s].x = am.x * ic;
    aMean[s].y = am.y * ic;
    aSelf[s] = selp[kp];
  }

  const int mrow0 = (wave << 4) + ((lane >> 4) << 3);
  constexpr int NT = OUTC / 16;

  if (((wave << 4) + 16) <= n_nodes) {  // uniform: full 16-row tile (hot path)
#pragma unroll
    for (int nt = 0; nt < NT; ++nt) {
      const int n = nt * 16 + (lane & 15);
      v8f c = {};
#pragma unroll
      for (int s = 0; s < KS; ++s) {
        const int k = s * 4 + kb;
        v2f b;
        b.x = wl[(long long)k * OUTC + n];
        b.y = wl[(long long)(k + 1) * OUTC + n];
        c = __builtin_amdgcn_wmma_f32_16x16x4_f32(false, aMean[s], false, b,
                                                  (short)0, c, false, false);
      }
#pragma unroll
      for (int s = 0; s < KS; ++s) {
        const int k = s * 4 + kb;
        v2f b;
        b.x = wr[(long long)k * OUTC + n];
        b.y = wr[(long long)(k + 1) * OUTC + n];
        c = __builtin_amdgcn_wmma_f32_16x16x4_f32(false, aSelf[s], false, b,
                                                  (short)0, c, false, false);
      }
      const float bi = bias[n];
#pragma unroll
      for (int r = 0; r < 8; ++r)
        out[(long long)(mrow0 + r) * OUTC + n] = fmaxf(c[r] + bi, 0.0f);
    }
  } else {  // rare ragged tail wave
#pragma unroll
    for (int nt = 0; nt < NT; ++nt) {
      const int n = nt * 16 + (lane & 15);
      v8f c = {};
#pragma unroll
      for (int s = 0; s < KS; ++s) {
        const int k = s * 4 + kb;
        v2f b;
        b.x = wl[(long long)k * OUTC + n];
        b.y = wl[(long long)(k + 1) * OUTC + n];
        c = __builtin_amdgcn_wmma_f32_16x16x4_f32(false, aMean[s], false, b,
                                                  (short)0, c, false, false);
      }
#pragma unroll
      for (int s = 0; s < KS; ++s) {
        const int k = s * 4 + kb;
        v2f b;
        b.x = wr[(long long)k * OUTC + n];
        b.y = wr[(long long)(k + 1) * OUTC + n];
        c = __builtin_amdgcn_wmma_f32_16x16x4_f32(false, aSelf[s], false, b,
                                                  (short)0, c, false, false);
      }
      const float bi = bias[n];
#pragma unroll
      for (int r = 0; r < 8; ++r)
        if (mrow0 + r < n_nodes)
          out[(long long)(mrow0 + r) * OUTC + n] = fmaxf(c[r] + bi, 0.0f);
    }
  }
}

// ---------------------------------------------------------------------------
// FC: out = h2 @ fc_w + fc_b   (h2: [N,32], fc_w: [32,nout], nout=1000).
// fc_w is staged once per block into LDS *pre-swizzled into the WMMA per-lane
// operand layout*: bsw[(nt*8+s)*32 + lane] = {B[k][n], B[k+1][n]} with
// k = 4s + (lane>>4)*2, n = nt*16 + (lane&15), ragged columns zero-padded.
// Inner loop B access = 8 x ds_load_b64 off ONE base VGPR with immediate
// offsets s*256B (lane l hits banks 2l,2l+1 -> conflict-free, no addr ALU).
// bias is staged zero-padded in LDS too. The A tile (16x32) stays resident in
// 8 v2f VGPRs across all 63 N-tiles, so the 400MB output store stream is the
// only heavy memory stream. Main loop over full tiles is branch-free; the one
// partial N-tile and ragged row waves are peeled out.
// ---------------------------------------------------------------------------
__global__ __launch_bounds__(256) void fc_wmma(
    const float* __restrict__ h, const float* __restrict__ w,
    const float* __restrict__ bias, float* __restrict__ out, int n_nodes,
    int nout, int npad) {
  extern __shared__ float lds[];  // [ntiles*512 + npad] floats
  const int ntiles = npad >> 4;
  v2f* bsw = (v2f*)lds;                       // ntiles*8*32 float2
  float* bbias = lds + (size_t)ntiles * 512;  // npad floats

  // ---- cooperative swizzled fill ----
  const int tid = threadIdx.x;
  const int total = ntiles * 256;  // (nt, s, lane) triples
  for (int idx = tid; idx < total; idx += 256) {
    const int fl = idx & 31;
    const int fs = (idx >> 5) & 7;
    const int fnt = idx >> 8;
    const int fk = fs * 4 + ((fl >> 4) << 1);
    const int fn = fnt * 16 + (fl & 15);
    v2f b = {0.0f, 0.0f};
    if (fn < nout) {
      b.x = w[(long long)fk * nout + fn];
      b.y = w[(long long)(fk + 1) * nout + fn];
    }
    bsw[idx] = b;
  }
  for (int cix = tid; cix < npad; cix += 256)
    bbias[cix] = (cix < nout) ? bias[cix] : 0.0f;
  __syncthreads();

  const int lane = tid & 31;
  const int wave = (blockIdx.x * blockDim.x + tid) >> 5;
  const int mtiles = (n_nodes + 15) >> 4;
  if (wave >= mtiles) return;  // uniform per wave

  const int m = (wave << 4) + (lane & 15);
  const int mc = (m < n_nodes) ? m : (n_nodes - 1);
  const int kb = (lane >> 4) << 1;

  const v2f* hp = (const v2f*)(h + (long long)mc * 32);
  v2f a[8];
#pragma unroll
  for (int s = 0; s < 8; ++s) a[s] = hp[s * 2 + (kb >> 1)];

  const int mrow0 = (wave << 4) + ((lane >> 4) << 3);
  const int nfull = nout >> 4;  // tiles with all 16 columns valid

  if (((wave << 4) + 16) <= n_nodes) {  // uniform: full-row hot path
    for (int nt = 0; nt < nfull; ++nt) {
      const v2f* bt = bsw + nt * 256 + lane;  // s stride = 32 v2f = 256B imm
      v8f c = {};
#pragma unroll
      for (int s = 0; s < 8; ++s)
        c = __builtin_amdgcn_wmma_f32_16x16x4_f32(false, a[s], false, bt[s * 32],
                                                  (short)0, c, false, false);
      const int n = nt * 16 + (lane & 15);
      const float bi = bbias[n];
#pragma unroll
      for (int r = 0; r < 8; ++r)
        out[(long long)(mrow0 + r) * nout + n] = c[r] + bi;
    }
    if (nfull < ntiles) {  // single partial N-tile: guard stores only
      const v2f* bt = bsw + nfull * 256 + lane;
      v8f c = {};
#pragma unroll
      for (int s = 0; s < 8; ++s)
        c = __builtin_amdgcn_wmma_f32_16x16x4_f32(false, a[s], false, bt[s * 32],
                                                  (short)0, c, false, false);
      const int n = nfull * 16 + (lane & 15);
      if (n < nout) {
        const float bi = bbias[n];
#pragma unroll
        for (int r = 0; r < 8; ++r)
          out[(long long)(mrow0 + r) * nout + n] = c[r] + bi;
      }
    }
  } else {  // rare ragged-row wave: fully guarded
    for (int nt = 0; nt < ntiles; ++nt) {
      const v2f* bt = bsw + nt * 256 + lane;
      v8f c = {};
#pragma unroll
      for (int s = 0; s < 8; ++s)
        c = __builtin_amdgcn_wmma_f32_16x16x4_f32(false, a[s], false, bt[s * 32],
                                                  (short)0, c, false, false);
      const int n = nt * 16 + (lane & 15);
      if (n < nout) {
        const float bi = bbias[n];
#pragma unroll
        for (int r = 0; r < 8; ++r)
          if (mrow0 + r < n_nodes)
            out[(long long)(mrow0 + r) * nout + n] = c[r] + bi;
      }
    }
  }
}

// ---------------------------------------------------------------------------
// Host-side pipeline (graph-capture safe: only kernel launches on `stream`).
// ---------------------------------------------------------------------------
extern "C" void kernel_launch(void* const* d_in, const int* in_sizes, int n_in,
                              void* d_out, int out_size, void* d_ws,
                              size_t ws_size, hipStream_t stream) {
  const float* x        = (const float*)d_in[0];
  const long long* ei   = (const long long*)d_in[1];  // int64 edge_index [2,E]
  const float* w1_l     = (const float*)d_in[2];
  const float* w1_r     = (const float*)d_in[3];
  const float* b1       = (const float*)d_in[4];
  const float* w2_l     = (const float*)d_in[5];
  const float* w2_r     = (const float*)d_in[6];
  const float* b2       = (const float*)d_in[7];
  const float* fc_w     = (const float*)d_in[8];
  const float* fc_b     = (const float*)d_in[9];
  float* out = (float*)d_out;

  const int n = in_sizes[0] / 16;       // 100000 nodes
  const long long E = in_sizes[1] / 2;  // 3.2M edges
  const int nout = in_sizes[9];         // 1000 (fc_b length)
  const int npad = ((nout + 15) / 16) * 16;  // 1008

  // Workspace layout (floats): agg1 | agg2 | cnt | invc | h1 | h2  (~45.6MB)
  float* agg1 = (float*)d_ws;
  float* agg2 = agg1 + (size_t)n * 16;
  float* cnt  = agg2 + (size_t)n * 32;
  float* invc = cnt + n;
  float* h1   = invc + n;
  float* h2   = h1 + (size_t)n * 32;

  // 1) zero agg1|agg2|cnt (contiguous: n*(16+32+1) floats)
  zero_f32<<<2048, 256, 0, stream>>>(agg1, (long long)n * 49);

  const int eb = (int)((E + 255) / 256);
  const long long* src = ei;
  const long long* dst = ei + E;

  // 2) layer-1 scatter (degree + features)
  scatter_deg_feat16<<<eb, 256, 0, stream>>>(x, src, dst, agg1, cnt, E);

  // 3) 1 / max(count, 1)
  inv_count<<<(n + 255) / 256, 256, 0, stream>>>(cnt, invc, n);

  const int mtiles = (n + 15) / 16;            // 6250
  const int lb = (mtiles * 32 + 255) / 256;    // 1 wave per M-tile, 8 waves/blk

  // 4) SAGE layer 1 (K=16 -> HIDDEN=32), relu
  sage_layer_wmma<16, 32><<<lb, 256, 0, stream>>>(agg1, invc, x, w1_l, w1_r,
                                                  b1, h1, n);
  // 5) layer-2 scatter
  scatter_feat32<<<eb, 256, 0, stream>>>(h1, src, dst, agg2, E);

  // 6) SAGE layer 2 (K=32 -> 32), relu
  sage_layer_wmma<32, 32><<<lb, 256, 0, stream>>>(agg2, invc, h1, w2_l, w2_r,
                                                  b2, h2, n);
  // 7) FC head (32 -> 1000) + bias; swizzled fc_w + bias staged in LDS
  const int ntiles = npad / 16;
  const size_t shbytes = ((size_t)ntiles * 512 + npad) * sizeof(float); // ~130KB
  fc_wmma<<<lb, 256, shbytes, stream>>>(h2, fc_w, fc_b, out, n, nout, npad);
}